// OptimalTransport_41755672052330
// MI455X (gfx1250) — compile-verified
//
#include <hip/hip_runtime.h>
#include <math.h>

// Problem constants (match reference)
constexpr int Bsz = 8;
constexpr int N   = 2048;
constexpr int D   = 256;
constexpr int M2  = 2049;               // N + 1 (with dust bin)
constexpr int ITERS = 20;
constexpr float LOG4096 = 8.31776616672f;   // log(4096) = -norm
constexpr float LOG2048 = 7.62461898944f;   // log(2048)
constexpr float INV_SCALE = 0.0625f;        // 1/sqrt(256)

typedef float v2f __attribute__((ext_vector_type(2)));
typedef float v8f __attribute__((ext_vector_type(8)));

// ---------------------------------------------------------------------------
// 1) Gram-matrix GEMM via f32 WMMA:  Z[b,n,m] = sum_d X[b,d,n]*X[b,d,m] / 16
//    2x2 register blocking: one wave owns a 32x32 tile (4 accumulators).
//    Per K-step (K=4): 2 A frags + 2 B frags (8 dword loads) feed 4 WMMAs.
//    A (16x4, f32): lanes 0-15 hold M=0..15; VGPR0 = K {0|2}, VGPR1 = K {1|3}.
//    B (4x16, f32): same K striping across lane halves, N = lane&15.
// ---------------------------------------------------------------------------
__global__ void ot_gemm_couplings(const float* __restrict__ X, float* __restrict__ Z) {
    const int b    = blockIdx.z;
    const int lane = threadIdx.x;           // 0..31  (wave32)
    const int wave = threadIdx.y;           // 0..3
    const int n0   = blockIdx.x * 32;
    const int m0   = (blockIdx.y * 4 + wave) * 32;

    const float* Xb = X + (size_t)b * D * N;
    const int lane15 = lane & 15;
    const int khalf  = (lane >> 4) * 2;     // lanes 16-31 carry K+2

    v8f acc00 = {0.f,0.f,0.f,0.f,0.f,0.f,0.f,0.f};
    v8f acc01 = acc00, acc10 = acc00, acc11 = acc00;

    #pragma unroll 2
    for (int k0 = 0; k0 < D; k0 += 4) {
        const int kk = k0 + khalf;
        const float* r0 = Xb + (size_t)kk * N;
        const float* r1 = Xb + (size_t)(kk + 1) * N;
        v2f a0, a1, b0, b1;
        a0.x = r0[n0 + lane15];      a0.y = r1[n0 + lane15];
        a1.x = r0[n0 + 16 + lane15]; a1.y = r1[n0 + 16 + lane15];
        b0.x = r0[m0 + lane15];      b0.y = r1[m0 + lane15];
        b1.x = r0[m0 + 16 + lane15]; b1.y = r1[m0 + 16 + lane15];
        acc00 = __builtin_amdgcn_wmma_f32_16x16x4_f32(false, a0, false, b0, (short)0, acc00, false, false);
        acc01 = __builtin_amdgcn_wmma_f32_16x16x4_f32(false, a0, false, b1, (short)0, acc01, false, false);
        acc10 = __builtin_amdgcn_wmma_f32_16x16x4_f32(false, a1, false, b0, (short)0, acc10, false, false);
        acc11 = __builtin_amdgcn_wmma_f32_16x16x4_f32(false, a1, false, b1, (short)0, acc11, false, false);
    }

    // C/D layout: VGPR r -> M = r (lanes 0-15) or r+8 (lanes 16-31), N = lane&15
    float* Zb = Z + (size_t)b * M2 * M2;
    const int rowadd = (lane >> 4) * 8;
    #pragma unroll
    for (int r = 0; r < 8; ++r) {
        const int row0 = n0 + r + rowadd;
        const int row1 = row0 + 16;
        const int col0 = m0 + lane15;
        const int col1 = col0 + 16;
        float v00 = acc00[r] * INV_SCALE;
        float v01 = acc01[r] * INV_SCALE;
        float v10 = acc10[r] * INV_SCALE;
        float v11 = acc11[r] * INV_SCALE;
        if (row0 == col0) v00 = -INFINITY;   // diagonal -> -inf
        if (row0 == col1) v01 = -INFINITY;
        if (row1 == col0) v10 = -INFINITY;
        if (row1 == col1) v11 = -INFINITY;
        Zb[(size_t)row0 * M2 + col0] = v00;
        Zb[(size_t)row0 * M2 + col1] = v01;
        Zb[(size_t)row1 * M2 + col0] = v10;
        Zb[(size_t)row1 * M2 + col1] = v11;
    }
}

// ---------------------------------------------------------------------------
// 2) Fill dust-bin row/column/corner with alpha
// ---------------------------------------------------------------------------
__global__ void ot_fill_bins(float* __restrict__ Z, const float* __restrict__ alpha_p) {
    const int idx = blockIdx.x * blockDim.x + threadIdx.x;
    if (idx >= Bsz * M2) return;
    const int b = idx / M2;
    const int j = idx - b * M2;
    const float a = alpha_p[0];
    float* Zb = Z + (size_t)b * M2 * M2;
    Zb[(size_t)N * M2 + j] = a;   // last row (includes corner)
    Zb[(size_t)j * M2 + N] = a;   // last column
}

__global__ void ot_zero_uv(float* __restrict__ uv) {
    const int idx = blockIdx.x * blockDim.x + threadIdx.x;
    if (idx < 2 * Bsz * M2) uv[idx] = 0.f;
}

// Online-logsumexp accumulate, -inf safe
__device__ __forceinline__ void lse_acc(float a, float& mx, float& s) {
    if (a > mx) {
        s = s * __expf(mx - a) + 1.0f;      // exp(-inf - finite) = 0 : safe
        mx = a;
    } else if (mx > -INFINITY) {
        s += __expf(a - mx);                // a may be -inf: exp(-inf)=0
    }                                       // both -inf: skip (avoid NaN)
}
__device__ __forceinline__ void lse_combine(float omx, float os, float& mx, float& s) {
    if (omx > mx) {
        s = os + s * __expf(mx - omx);
        mx = omx;
    } else if (omx > -INFINITY) {
        s += os * __expf(omx - mx);
    }
}

__device__ __forceinline__ float log_mu_nu(int i) {   // identical for mu and nu here
    return (i < N) ? -LOG4096 : (LOG2048 - LOG4096);
}

// ---------------------------------------------------------------------------
// 3a) u[n] = log_mu[n] - logsumexp_m( Z[n,m] + v[m] )   (one 256-thread block per row)
// ---------------------------------------------------------------------------
__global__ void ot_row_lse(const float* __restrict__ Z, const float* __restrict__ v,
                           float* __restrict__ u) {
    const int n = blockIdx.x;
    const int b = blockIdx.y;
    const int tid = threadIdx.x;
    const float* Zrow = Z + (size_t)b * M2 * M2 + (size_t)n * M2;
    const float* vb   = v + b * M2;

    float mx = -INFINITY, s = 0.f;
    for (int m = tid; m < M2; m += 256)
        lse_acc(Zrow[m] + vb[m], mx, s);

    // wave32 reduce
    #pragma unroll
    for (int off = 16; off > 0; off >>= 1) {
        const float omx = __shfl_xor(mx, off, 32);
        const float os  = __shfl_xor(s,  off, 32);
        lse_combine(omx, os, mx, s);
    }
    __shared__ float smx[8], ss[8];
    if ((tid & 31) == 0) { smx[tid >> 5] = mx; ss[tid >> 5] = s; }
    __syncthreads();
    if (tid == 0) {
        mx = smx[0]; s = ss[0];
        #pragma unroll
        for (int w = 1; w < 8; ++w) lse_combine(smx[w], ss[w], mx, s);
        u[b * M2 + n] = log_mu_nu(n) - (mx + __logf(s));
    }
}

// ---------------------------------------------------------------------------
// 3b) v[m] = log_nu[m] - logsumexp_n( Z[n,m] + u[n] )
//     Block = (64 columns) x (8 row-slices), 512 threads / 16 waves:
//     slice y accumulates n = y, y+8, ... ; LDS-reduce 8 partials per column.
//     8x more parallelism than one-thread-per-column; fully coalesced reads;
//     u[n] reads are wave-uniform (broadcast).
// ---------------------------------------------------------------------------
__global__ void ot_col_lse(const float* __restrict__ Z, const float* __restrict__ u,
                           float* __restrict__ v) {
    const int b  = blockIdx.y;
    const int x  = threadIdx.x;              // 0..63 : column within chunk
    const int y  = threadIdx.y;              // 0..7  : row slice
    const int m  = blockIdx.x * 64 + x;
    const bool live = (m < M2);

    const float* Zb = Z + (size_t)b * M2 * M2;
    const float* ub = u + b * M2;

    float mx = -INFINITY, s = 0.f;
    if (live) {
        for (int n = y; n < M2; n += 8)
            lse_acc(Zb[(size_t)n * M2 + m] + ub[n], mx, s);
    }

    __shared__ float smx[8][64];
    __shared__ float ss[8][64];
    smx[y][x] = mx; ss[y][x] = s;
    __syncthreads();

    if (y == 0 && live) {
        mx = smx[0][x]; s = ss[0][x];
        #pragma unroll
        for (int yy = 1; yy < 8; ++yy)
            lse_combine(smx[yy][x], ss[yy][x], mx, s);
        v[b * M2 + m] = log_mu_nu(m) - (mx + __logf(s));
    }
}

// ---------------------------------------------------------------------------
// 4) S = exp(Z + u + v + log(4096)) * spatial_mask   (in place over d_out)
//    positions: (B, N, 2) with [:,0]=y, [:,1]=x ; mask iff |dx|<=0.1 & |dy|<=0.1, diag off
// ---------------------------------------------------------------------------
__global__ void ot_finalize(float* __restrict__ Z, const float* __restrict__ u,
                            const float* __restrict__ v, const float* __restrict__ pos) {
    const int b = blockIdx.z;
    const int n = blockIdx.y;
    const int m = blockIdx.x * 256 + threadIdx.x;
    if (m >= M2) return;

    float* Zb = Z + (size_t)b * M2 * M2;
    const float zv = Zb[(size_t)n * M2 + m];
    float e = __expf(zv + u[b * M2 + n] + v[b * M2 + m] + LOG4096);

    if (n < N && m < N) {
        const float yn = pos[((size_t)b * N + n) * 2 + 0];
        const float xn = pos[((size_t)b * N + n) * 2 + 1];
        const float ym = pos[((size_t)b * N + m) * 2 + 0];
        const float xm = pos[((size_t)b * N + m) * 2 + 1];
        const bool adj = (n != m) && (fabsf(xn - xm) <= 0.1f) && (fabsf(yn - ym) <= 0.1f);
        if (!adj) e = 0.f;
    }
    Zb[(size_t)n * M2 + m] = e;
}

// ---------------------------------------------------------------------------
extern "C" void kernel_launch(void* const* d_in, const int* in_sizes, int n_in,
                              void* d_out, int out_size, void* d_ws, size_t ws_size,
                              hipStream_t stream) {
    const float* X     = (const float*)d_in[0];   // (B, D, N) f32
    const float* pos   = (const float*)d_in[1];   // (B, N, 2) f32
    // d_in[2] = active_nodes (unused by reference)
    const float* alpha = (const float*)d_in[3];   // (1,) f32

    float* Z = (float*)d_out;                     // (B, 2049, 2049) couplings -> S
    float* u = (float*)d_ws;                      // (B, 2049)
    float* v = u + Bsz * M2;                      // (B, 2049)

    // 1) couplings via WMMA GEMM (32x32 tile per wave) + bins + u/v init
    ot_gemm_couplings<<<dim3(N / 32, N / 128, Bsz), dim3(32, 4), 0, stream>>>(X, Z);
    ot_fill_bins<<<(Bsz * M2 + 255) / 256, 256, 0, stream>>>(Z, alpha);
    ot_zero_uv<<<(2 * Bsz * M2 + 255) / 256, 256, 0, stream>>>(u);

    // 2) 20 Sinkhorn iterations (row LSE -> u, col LSE -> v)
    const int cchunks = (M2 + 63) / 64;           // 33
    for (int it = 0; it < ITERS; ++it) {
        ot_row_lse<<<dim3(M2, Bsz), 256, 0, stream>>>(Z, v, u);
        ot_col_lse<<<dim3(cchunks, Bsz), dim3(64, 8), 0, stream>>>(Z, u, v);
    }

    // 3) exp + scaling + spatial mask, in place
    const int mchunks = (M2 + 255) / 256;         // 9
    ot_finalize<<<dim3(mchunks, M2, Bsz), 256, 0, stream>>>(Z, u, v, pos);
}